// GatedGATLayer_50500225466440
// MI455X (gfx1250) — compile-verified
//
#include <hip/hip_runtime.h>
#include <math.h>

// ---------------- problem constants ----------------
#define IN_DIM  64
#define OUT_DIM 64
#define HEADS   4
#define HC      256   // HEADS*OUT_DIM
#define EDGE_DIM 2

typedef __attribute__((ext_vector_type(2))) float v2f;
typedef __attribute__((ext_vector_type(8))) float v8f;

// ---------------- init: zero accumulators, -inf for segment-max ----------------
__global__ void k_init(float* __restrict__ z0,    // 3N floats (deg + loop_attr)
                       float* __restrict__ agg,   // 256N floats
                       float* __restrict__ denom, // 4N floats
                       float* __restrict__ amax,  // 4N floats
                       int n) {
    long long t = (long long)blockIdx.x * blockDim.x + threadIdx.x;
    long long n3 = 3LL * n, n256 = 256LL * n, n4 = 4LL * n;
    if (t < n3)        { z0[t] = 0.0f; return; }
    t -= n3;
    if (t < n256)      { agg[t] = 0.0f; return; }
    t -= n256;
    if (t < n4)        { denom[t] = 0.0f; return; }
    t -= n4;
    if (t < n4)        { amax[t] = -INFINITY; }
}

// ---------------- degree + loop_attr sums ----------------
__global__ void k_deg(const int* __restrict__ dst, const float* __restrict__ eattr,
                      float* __restrict__ deg, float* __restrict__ loopa, int E) {
    int e = blockIdx.x * blockDim.x + threadIdx.x;
    if (e >= E) return;
    int d = dst[e];
    float2 ea = ((const float2*)eattr)[e];
    atomicAdd(&deg[d], 1.0f);
    atomicAdd(&loopa[2 * d + 0], ea.x);
    atomicAdd(&loopa[2 * d + 1], ea.y);
}

__global__ void k_loopdiv(const float* __restrict__ deg, float* __restrict__ loopa, int n) {
    int i = blockIdx.x * blockDim.x + threadIdx.x;
    if (i >= n) return;
    float den = fmaxf(deg[i], 1.0f);
    loopa[2 * i + 0] /= den;
    loopa[2 * i + 1] /= den;
}

// ---------------- node GEMMs: xl = x@Wl + bl, xr = x@Wr + br  (WMMA fp32) ----------------
// one wave computes a 16x64 slab (4 N-tiles share one A fragment);
// K=64 -> 16 steps x 4 chained v_wmma_f32_16x16x4_f32
__global__ void k_gemm_node(const float* __restrict__ x,
                            const float* __restrict__ Wl, const float* __restrict__ bl,
                            const float* __restrict__ Wr, const float* __restrict__ br,
                            float* __restrict__ xl, float* __restrict__ xr, int n) {
    const float* W    = blockIdx.z ? Wr : Wl;
    const float* bias = blockIdx.z ? br : bl;
    float*       out  = blockIdx.z ? xr : xl;

    unsigned lane = threadIdx.x;            // 0..31
    int mtile = blockIdx.x * 16;
    int nbase = blockIdx.y * 64;            // 4 consecutive 16-wide tiles
    int rowA  = mtile + (lane & 15);
    if (rowA >= n) rowA = n - 1;            // clamp (EXEC must stay all-ones)
    int colB  = lane & 15;
    int khalf = (lane >> 4) * 2;            // lanes 0-15: K k..k+1, lanes 16-31: K k+2..k+3

    v8f c0 = {}, c1 = {}, c2 = {}, c3 = {};
#pragma unroll
    for (int k = 0; k < IN_DIM; k += 4) {
        v2f a;
        a.x = x[(long long)rowA * IN_DIM + k + khalf];
        a.y = x[(long long)rowA * IN_DIM + k + khalf + 1];
        const float* w0 = W + (long long)(k + khalf) * HC + nbase + colB;
        const float* w1 = W + (long long)(k + khalf + 1) * HC + nbase + colB;
        v2f b0, b1, b2, b3;
        b0.x = w0[0];   b0.y = w1[0];
        b1.x = w0[16];  b1.y = w1[16];
        b2.x = w0[32];  b2.y = w1[32];
        b3.x = w0[48];  b3.y = w1[48];
        c0 = __builtin_amdgcn_wmma_f32_16x16x4_f32(false, a, false, b0, (short)0, c0, false, false);
        c1 = __builtin_amdgcn_wmma_f32_16x16x4_f32(false, a, false, b1, (short)0, c1, false, false);
        c2 = __builtin_amdgcn_wmma_f32_16x16x4_f32(false, a, false, b2, (short)0, c2, false, false);
        c3 = __builtin_amdgcn_wmma_f32_16x16x4_f32(false, a, false, b3, (short)0, c3, false, false);
    }
    v8f acc[4] = {c0, c1, c2, c3};
#pragma unroll
    for (int t = 0; t < 4; ++t) {
        int col = nbase + t * 16 + colB;
        float bc = bias[col];
#pragma unroll
        for (int i = 0; i < 8; ++i) {
            int r = mtile + i + 8 * (lane >> 4);
            if (r < n) out[(long long)r * HC + col] = acc[t][i] + bc;
        }
    }
}

// ---------------- per-edge attention score + segment max ----------------
__device__ __forceinline__ void atomicMaxF(float* addr, float v) {
    if (v >= 0.0f) atomicMax((int*)addr, __float_as_int(v));
    else           atomicMin((unsigned int*)addr, __float_as_uint(v));
}

__device__ __forceinline__ float dot_lrelu4(float4 a, float4 r, float4 w02, float4 w13,
                                            float e0, float e1, float4 t) {
    float m, acc = 0.0f;
    m = a.x + r.x + e0 * w02.x + e1 * w13.x; m = (m > 0.0f) ? m : 0.2f * m; acc += m * t.x;
    m = a.y + r.y + e0 * w02.y + e1 * w13.y; m = (m > 0.0f) ? m : 0.2f * m; acc += m * t.y;
    m = a.z + r.z + e0 * w02.z + e1 * w13.z; m = (m > 0.0f) ? m : 0.2f * m; acc += m * t.z;
    m = a.w + r.w + e0 * w02.w + e1 * w13.w; m = (m > 0.0f) ? m : 0.2f * m; acc += m * t.w;
    return acc;
}

__global__ void k_alpha(const int* __restrict__ src, const int* __restrict__ dst,
                        const float* __restrict__ eattr, const float* __restrict__ loopa,
                        const float* __restrict__ xl, const float* __restrict__ xr,
                        const float* __restrict__ We, const float* __restrict__ att,
                        float* __restrict__ alpha, float* __restrict__ amax,
                        int E, int E2) {
    int wave = threadIdx.x >> 5;
    int eid  = blockIdx.x * (blockDim.x >> 5) + wave;
    if (eid >= E2) return;
    unsigned lane = threadIdx.x & 31;

    int s, d; float e0, e1;
    if (eid < E) { s = src[eid]; d = dst[eid];
                   float2 ea = ((const float2*)eattr)[eid]; e0 = ea.x; e1 = ea.y; }
    else         { s = d = eid - E;
                   float2 ea = ((const float2*)loopa)[s];   e0 = ea.x; e1 = ea.y; }

    // lane handles channels [lane*8, lane*8+8): two contiguous float4 loads per tensor
    const float4* xl4  = (const float4*)xl;
    const float4* xr4  = (const float4*)xr;
    const float4* We4  = (const float4*)We;
    const float4* att4 = (const float4*)att;
    long long sb = (long long)s * 64 + lane * 2;
    long long db = (long long)d * 64 + lane * 2;
    float4 a0 = xl4[sb],  a1 = xl4[sb + 1];
    float4 r0 = xr4[db],  r1 = xr4[db + 1];
    float4 w00 = We4[lane * 2],      w01 = We4[lane * 2 + 1];       // We row 0
    float4 w10 = We4[64 + lane * 2], w11 = We4[64 + lane * 2 + 1];  // We row 1
    float4 t0 = att4[lane * 2],      t1 = att4[lane * 2 + 1];

    float acc = dot_lrelu4(a0, r0, w00, w10, e0, e1, t0)
              + dot_lrelu4(a1, r1, w01, w11, e0, e1, t1);

    acc += __shfl_xor(acc, 4);
    acc += __shfl_xor(acc, 2);
    acc += __shfl_xor(acc, 1);
    int head = lane >> 3;                     // 8 lanes per head
    if ((lane & 7) == 0) {
        alpha[(long long)eid * HEADS + head] = acc;
        atomicMaxF(&amax[(long long)d * HEADS + head], acc);
    }
}

// ---------------- exp + segment-sum denominator ----------------
__global__ void k_exp(const int* __restrict__ dst, const float* __restrict__ amax,
                      float* __restrict__ alpha, float* __restrict__ denom,
                      int E, int E2) {
    long long t = (long long)blockIdx.x * blockDim.x + threadIdx.x;
    if (t >= (long long)E2 * HEADS) return;
    int eid = (int)(t >> 2);
    int h   = (int)(t & 3);
    int d   = (eid < E) ? dst[eid] : eid - E;
    float ex = __expf(alpha[t] - amax[(long long)d * HEADS + h]);
    alpha[t] = ex;
    atomicAdd(&denom[(long long)d * HEADS + h], ex);
}

// ---------------- weighted scatter-add aggregation ----------------
__global__ void k_agg(const int* __restrict__ src, const int* __restrict__ dst,
                      const float* __restrict__ xl, const float* __restrict__ alpha,
                      const float* __restrict__ denom, float* __restrict__ agg,
                      int E, int E2) {
    int wave = threadIdx.x >> 5;
    int eid  = blockIdx.x * (blockDim.x >> 5) + wave;
    if (eid >= E2) return;
    unsigned lane = threadIdx.x & 31;

    int s, d;
    if (eid < E) { s = src[eid]; d = dst[eid]; }
    else         { s = d = eid - E; }

    int head = lane >> 3;
    float w = alpha[(long long)eid * HEADS + head] / denom[(long long)d * HEADS + head];

    const float4* xl4 = (const float4*)xl;
    long long sb = (long long)s * 64 + lane * 2;
    float4 v0 = xl4[sb], v1 = xl4[sb + 1];
    float* base = agg + (long long)d * HC + lane * 8;
    atomicAdd(base + 0, v0.x * w);
    atomicAdd(base + 1, v0.y * w);
    atomicAdd(base + 2, v0.z * w);
    atomicAdd(base + 3, v0.w * w);
    atomicAdd(base + 4, v1.x * w);
    atomicAdd(base + 5, v1.y * w);
    atomicAdd(base + 6, v1.z * w);
    atomicAdd(base + 7, v1.w * w);
}

// ---------------- output GEMM: out = elu((agg+bias_gat)@Wo + bo) + x  (WMMA fp32) ----------------
// one wave computes a full 16x64 output slab; K=256 -> 64 steps x 4 WMMAs
__global__ void k_out(const float* __restrict__ agg, const float* __restrict__ bias_gat,
                      const float* __restrict__ Wo, const float* __restrict__ bo,
                      const float* __restrict__ x, float* __restrict__ out, int n) {
    unsigned lane = threadIdx.x;
    int mtile = blockIdx.x * 16;
    int rowA  = mtile + (lane & 15);
    if (rowA >= n) rowA = n - 1;
    int colB  = lane & 15;
    int khalf = (lane >> 4) * 2;

    v8f c0 = {}, c1 = {}, c2 = {}, c3 = {};
#pragma unroll 8
    for (int k = 0; k < HC; k += 4) {
        v2f a;
        a.x = agg[(long long)rowA * HC + k + khalf]     + bias_gat[k + khalf];
        a.y = agg[(long long)rowA * HC + k + khalf + 1] + bias_gat[k + khalf + 1];
        const float* w0 = Wo + (long long)(k + khalf) * OUT_DIM + colB;
        const float* w1 = Wo + (long long)(k + khalf + 1) * OUT_DIM + colB;
        v2f b0, b1, b2, b3;
        b0.x = w0[0];   b0.y = w1[0];
        b1.x = w0[16];  b1.y = w1[16];
        b2.x = w0[32];  b2.y = w1[32];
        b3.x = w0[48];  b3.y = w1[48];
        c0 = __builtin_amdgcn_wmma_f32_16x16x4_f32(false, a, false, b0, (short)0, c0, false, false);
        c1 = __builtin_amdgcn_wmma_f32_16x16x4_f32(false, a, false, b1, (short)0, c1, false, false);
        c2 = __builtin_amdgcn_wmma_f32_16x16x4_f32(false, a, false, b2, (short)0, c2, false, false);
        c3 = __builtin_amdgcn_wmma_f32_16x16x4_f32(false, a, false, b3, (short)0, c3, false, false);
    }
    v8f acc[4] = {c0, c1, c2, c3};
#pragma unroll
    for (int t = 0; t < 4; ++t) {
        int col = t * 16 + colB;
        float bc = bo[col];
#pragma unroll
        for (int i = 0; i < 8; ++i) {
            int r = mtile + i + 8 * (lane >> 4);
            if (r < n) {
                float v = acc[t][i] + bc;
                v = (v > 0.0f) ? v : (__expf(v) - 1.0f);   // elu
                out[(long long)r * OUT_DIM + col] = v + x[(long long)r * OUT_DIM + col];
            }
        }
    }
}

// ---------------- host-side launcher ----------------
extern "C" void kernel_launch(void* const* d_in, const int* in_sizes, int n_in,
                              void* d_out, int out_size, void* d_ws, size_t ws_size,
                              hipStream_t stream) {
    const float* x        = (const float*)d_in[0];
    const int*   eidx     = (const int*)  d_in[1];   // [2, E] flat: row0=src, row1=dst
    const float* eattr    = (const float*)d_in[2];
    const float* Wl       = (const float*)d_in[3];
    const float* bl       = (const float*)d_in[4];
    const float* Wr       = (const float*)d_in[5];
    const float* br       = (const float*)d_in[6];
    const float* We       = (const float*)d_in[7];
    const float* att      = (const float*)d_in[8];
    const float* bias_gat = (const float*)d_in[9];
    const float* Wo       = (const float*)d_in[10];
    const float* bo       = (const float*)d_in[11];
    float* out = (float*)d_out;

    const int N  = in_sizes[0] / IN_DIM;
    const int E  = in_sizes[1] / 2;
    const int E2 = E + N;
    const int*   src = eidx;
    const int*   dst = eidx + E;

    // workspace layout (floats)
    float* ws    = (float*)d_ws;
    float* deg   = ws;                      // N
    float* loopa = ws + (long long)N;       // 2N
    float* xl    = ws + 3LL * N;            // 256N
    float* xr    = xl + 256LL * N;          // 256N
    float* agg   = xr + 256LL * N;          // 256N
    float* amax  = agg + 256LL * N;         // 4N
    float* denom = amax + 4LL * N;          // 4N
    float* alpha = denom + 4LL * N;         // 4*E2

    const int mtiles = (N + 15) / 16;       // 3125 for N=50000

    // 1) init accumulators
    {
        long long tot = 267LL * N;
        int blocks = (int)((tot + 255) / 256);
        k_init<<<blocks, 256, 0, stream>>>(deg, agg, denom, amax, N);
    }
    // 2) degree + loop-attr sums, then normalize
    k_deg<<<(E + 255) / 256, 256, 0, stream>>>(dst, eattr, deg, loopa, E);
    k_loopdiv<<<(N + 255) / 256, 256, 0, stream>>>(deg, loopa, N);
    // 3) xl / xr node GEMMs (WMMA, 16x64 slab per wave)
    {
        dim3 grid(mtiles, HC / 64, 2);
        k_gemm_node<<<grid, 32, 0, stream>>>(x, Wl, bl, Wr, br, xl, xr, N);
    }
    // 4) attention scores + segment max (one wave per edge, float4 gathers)
    k_alpha<<<(E2 + 7) / 8, 256, 0, stream>>>(src, dst, eattr, loopa, xl, xr, We, att,
                                              alpha, amax, E, E2);
    // 5) exp + denom
    {
        long long tot = (long long)E2 * HEADS;
        k_exp<<<(int)((tot + 255) / 256), 256, 0, stream>>>(dst, amax, alpha, denom, E, E2);
    }
    // 6) weighted scatter aggregation (one wave per edge, float4 gathers)
    k_agg<<<(E2 + 7) / 8, 256, 0, stream>>>(src, dst, xl, alpha, denom, agg, E, E2);
    // 7) output projection + ELU + residual (WMMA, 16x64 slab per wave)
    {
        dim3 grid(mtiles);
        k_out<<<grid, 32, 0, stream>>>(agg, bias_gat, Wo, bo, x, out, N);
    }
}